// Model_39676907883922
// MI455X (gfx1250) — compile-verified
//
#include <hip/hip_runtime.h>
#include <math.h>

typedef __attribute__((ext_vector_type(2))) float v2f;
typedef __attribute__((ext_vector_type(8))) float v8f;

#define S_DIM 4096
#define D_DIM 64
#define JTILES (S_DIM / 16)      // 256 j-tiles of width 16
#define KEEP_INV (1.0f / 0.9f)   // inverted-dropout scale, p = 0.1
#define SQRT_D 8.0f              // sqrt(64)

// ---------------------------------------------------------------------------
// Fragment helpers (wave32, V_WMMA_F32_16X16X4_F32)
// A 16x4 f32 layout: lane<16 -> M=lane, K = v + 0 ; lane>=16 -> M=lane-16, K = v + 2
// B 4x16  f32 layout: lane<16 -> N=lane, K = v + 0 ; lane>=16 -> N=lane-16, K = v + 2
// C/D 16x16 f32:      M = v + (lane>=16 ? 8 : 0), N = lane % 16
// ---------------------------------------------------------------------------

__device__ __forceinline__ void load_qA(v2f* qA, const float* __restrict__ Q,
                                        int i0, int lane) {
    const int koff = (lane >= 16) ? 2 : 0;
    const int m = lane & 15;
    const float* qrow = Q + (size_t)(i0 + m) * D_DIM;
#pragma unroll
    for (int s = 0; s < 16; ++s)
        qA[s] = *(const v2f*)(qrow + 4 * s + koff);
}

// QK^T 16x16 tile for rows [i0,i0+16), cols [j0,j0+16), scaled by sqrt(D).
__device__ __forceinline__ v8f qk_tile(const v2f* qA, const float* __restrict__ Kmat,
                                       int j0, int lane) {
    const int koff = (lane >= 16) ? 2 : 0;
    const int n = lane & 15;
    const float* krow = Kmat + (size_t)(j0 + n) * D_DIM;
    v8f c = {};
#pragma unroll
    for (int s = 0; s < 16; ++s) {
        v2f b = *(const v2f*)(krow + 4 * s + koff);
        c = __builtin_amdgcn_wmma_f32_16x16x4_f32(false, qA[s], false, b,
                                                  (short)0, c, false, false);
    }
#pragma unroll
    for (int v = 0; v < 8; ++v) c[v] *= SQRT_D;
    return c;
}

// ---------------------------------------------------------------------------
// Pass A: global sum of r = floor(t * row_rand).  1024 waves (256 row-tiles x
// 4 j-chunks), deterministic block partials (no float atomics).
// ---------------------------------------------------------------------------
__global__ __launch_bounds__(256)
void passA_kernel(const float* __restrict__ Q, const float* __restrict__ Kmat,
                  const float* __restrict__ Udrop, const float* __restrict__ rowRand,
                  float* __restrict__ partials) {
    __shared__ float wsum[8];
    const int lane = threadIdx.x & 31;
    const int wid  = threadIdx.x >> 5;
    const int gw   = blockIdx.x * 8 + wid;   // [0, 1024)
    const int rowtile = gw >> 2;
    const int chunk   = gw & 3;
    const int i0 = rowtile * 16;
    const int half16 = (lane >= 16) ? 8 : 0;
    const int n = lane & 15;

    v2f qA[16];
    load_qA(qA, Q, i0, lane);
    float rr[8];
#pragma unroll
    for (int v = 0; v < 8; ++v) rr[v] = rowRand[i0 + v + half16];

    float rsum = 0.0f;
    for (int jt = chunk * (JTILES / 4); jt < (chunk + 1) * (JTILES / 4); ++jt) {
        const int j0 = jt * 16;
        v8f c = qk_tile(qA, Kmat, j0, lane);
#pragma unroll
        for (int v = 0; v < 8; ++v) {
            const int gi = i0 + v + half16;
            float u = __builtin_nontemporal_load(&Udrop[(size_t)gi * S_DIM + j0 + n]);
            float t = (u >= 0.1f) ? c[v] * KEEP_INV : 0.0f;
            rsum += floorf(t * rr[v]);
        }
    }
#pragma unroll
    for (int off = 16; off; off >>= 1) rsum += __shfl_xor(rsum, off, 32);
    if (lane == 0) wsum[wid] = rsum;
    __syncthreads();
    if (threadIdx.x == 0) {
        float s = 0.0f;
#pragma unroll
        for (int w = 0; w < 8; ++w) s += wsum[w];
        partials[blockIdx.x] = s;
    }
}

__global__ void reduce_kernel(float* wsf, int n) {
    if (threadIdx.x == 0 && blockIdx.x == 0) {
        float s = 0.0f;
        for (int i = 0; i < n; ++i) s += wsf[i];
        wsf[128] = s;   // total slot
    }
}

// ---------------------------------------------------------------------------
// Pass B: acc = (t .* r) @ V, unnormalized, per j-chunk.  LDS re-fragment of
// the QK tile (stride-17 pad -> conflict free) feeds the second WMMA GEMM.
// ---------------------------------------------------------------------------
__global__ __launch_bounds__(256)
void passB_kernel(const float* __restrict__ Q, const float* __restrict__ Kmat,
                  const float* __restrict__ V, const float* __restrict__ Udrop,
                  const float* __restrict__ rowRand, float* __restrict__ acc,
                  int jtiles_per_chunk) {
    __shared__ float tile[8][16 * 17];
    const int lane = threadIdx.x & 31;
    const int wid  = threadIdx.x >> 5;
    const int gw   = blockIdx.x * 8 + wid;
    const int chunk   = gw >> 8;       // gw / 256
    const int rowtile = gw & 255;      // gw % 256
    const int i0 = rowtile * 16;
    const int half16 = (lane >= 16) ? 8 : 0;
    const int koff   = (lane >= 16) ? 2 : 0;
    const int n = lane & 15;
    float* myt = &tile[wid][0];

    v2f qA[16];
    load_qA(qA, Q, i0, lane);
    float rr[8];
#pragma unroll
    for (int v = 0; v < 8; ++v) rr[v] = rowRand[i0 + v + half16];

    v8f accv[4] = {v8f{}, v8f{}, v8f{}, v8f{}};

    const int jt_begin = chunk * jtiles_per_chunk;
    const int jt_end   = jt_begin + jtiles_per_chunk;
    for (int jt = jt_begin; jt < jt_end; ++jt) {
        const int j0 = jt * 16;
        v8f c = qk_tile(qA, Kmat, j0, lane);
        // elementwise: dropout, floor-mask, a = t * r
#pragma unroll
        for (int v = 0; v < 8; ++v) {
            const int gi = i0 + v + half16;
            float u = __builtin_nontemporal_load(&Udrop[(size_t)gi * S_DIM + j0 + n]);
            float t = (u >= 0.1f) ? c[v] * KEEP_INV : 0.0f;
            float r = floorf(t * rr[v]);
            c[v] = t * r;
        }
        // C/D layout -> LDS 16x16 tile (row stride 17)
#pragma unroll
        for (int v = 0; v < 8; ++v)
            myt[(v + half16) * 17 + n] = c[v];
        asm volatile("s_wait_dscnt 0x0" ::: "memory");
        // reload as A fragments for the second GEMM (K2 = 16 -> 4 k-steps)
        v2f a2[4];
#pragma unroll
        for (int s = 0; s < 4; ++s) {
            const int k0 = 4 * s + koff;
            a2[s].x = myt[n * 17 + k0];
            a2[s].y = myt[n * 17 + k0 + 1];
        }
        asm volatile("" ::: "memory");  // keep next iter's LDS writes after these reads
        // acc += A2(16x16) * V(16x64), split into 4 n-blocks of 16
#pragma unroll
        for (int s = 0; s < 4; ++s) {
            const float* vrow = V + (size_t)(j0 + 4 * s + koff) * D_DIM;
#pragma unroll
            for (int nb = 0; nb < 4; ++nb) {
                v2f b;
                b.x = vrow[nb * 16 + n];
                b.y = vrow[D_DIM + nb * 16 + n];
                accv[nb] = __builtin_amdgcn_wmma_f32_16x16x4_f32(
                    false, a2[s], false, b, (short)0, accv[nb], false, false);
            }
        }
    }

    float* obase = acc + (size_t)chunk * S_DIM * D_DIM;
#pragma unroll
    for (int nb = 0; nb < 4; ++nb)
#pragma unroll
        for (int v = 0; v < 8; ++v)
            obase[(size_t)(i0 + v + half16) * D_DIM + nb * 16 + n] = accv[nb][v];
}

// ---------------------------------------------------------------------------
// Finalize: out = (sum over chunks) / total_r.  acc==nullptr -> in-place scale.
// ---------------------------------------------------------------------------
__global__ __launch_bounds__(256)
void finalize_kernel(float* __restrict__ out, const float* __restrict__ acc,
                     int nchunks, const float* __restrict__ totalp) {
    const int idx = blockIdx.x * blockDim.x + threadIdx.x;
    if (idx >= S_DIM * D_DIM) return;
    const float inv = 1.0f / *totalp;
    float s;
    if (acc) {
        s = 0.0f;
        for (int c = 0; c < nchunks; ++c)
            s += acc[(size_t)c * S_DIM * D_DIM + idx];
    } else {
        s = out[idx];
    }
    out[idx] = s * inv;
}

// ---------------------------------------------------------------------------
extern "C" void kernel_launch(void* const* d_in, const int* in_sizes, int n_in,
                              void* d_out, int out_size, void* d_ws, size_t ws_size,
                              hipStream_t stream) {
    const float* Q    = (const float*)d_in[0];
    const float* Kmat = (const float*)d_in[1];
    const float* V    = (const float*)d_in[2];
    const float* U    = (const float*)d_in[3];
    const float* rr   = (const float*)d_in[4];
    // d_in[5] ("x") is unused by the reference.
    float* out = (float*)d_out;
    float* wsf = (float*)d_ws;

    // ws layout (floats): [0..127] passA partials, [128] total, [256..] acc chunks
    const size_t hdr = 256;
    const size_t ws_floats = ws_size / sizeof(float);
    const size_t avail = (ws_floats > hdr) ? ws_floats - hdr : 0;
    const size_t od = (size_t)S_DIM * D_DIM;
    int JC; bool useWs;
    if      (avail >= 4 * od) { JC = 4; useWs = true;  }
    else if (avail >= 2 * od) { JC = 2; useWs = true;  }
    else if (avail >= 1 * od) { JC = 1; useWs = true;  }
    else                      { JC = 1; useWs = false; }   // accumulate into d_out

    passA_kernel<<<128, 256, 0, stream>>>(Q, Kmat, U, rr, wsf);
    reduce_kernel<<<1, 1, 0, stream>>>(wsf, 128);
    float* accp = useWs ? (wsf + hdr) : out;
    passB_kernel<<<32 * JC, 256, 0, stream>>>(Q, Kmat, V, U, rr, accp, JTILES / JC);
    finalize_kernel<<<(S_DIM * D_DIM + 255) / 256, 256, 0, stream>>>(
        out, useWs ? (wsf + hdr) : nullptr, JC, wsf + 128);
}